// PolicyLP_Plus_66477503808088
// MI455X (gfx1250) — compile-verified
//
#include <hip/hip_runtime.h>

#define NN 25
#define NV 625          // N*N variables
#define MP 64           // padded constraint count (real m = 50)
#define MST 65          // LDS row stride for M (65 ≡ 1 mod 64 banks -> conflict-free columns)
#define EPSQ 0.1f
#define NSTEPS 48

typedef float v2f __attribute__((ext_vector_type(2)));
typedef float v8f __attribute__((ext_vector_type(8)));

__global__ __launch_bounds__(256)
void policy_lp_kernel(const float* __restrict__ xin,
                      const float* __restrict__ gin,
                      const float* __restrict__ vin,
                      float* __restrict__ out)
{
  __shared__ float sM[MP][MST];
  __shared__ float sz[NV], sp[NV], sw[NV], sd[NV], sg[NV], sdx[NV];
  __shared__ float rowsum[NN], colsum[NN], rwv[NN], cwv[NN], rrv[NN], rcv[NN];
  __shared__ float snu[MP];
  __shared__ float red[512];
  __shared__ float scal[8];

  const int b    = blockIdx.x;
  const int tid  = threadIdx.x;
  const int lane = tid & 31;
  const int wave = tid >> 5;

  // ---------- setup: f, w = f_i * v_j, p = -scores, z0 = 1/N ----------
  if (tid == 0) {
    float s1 = 0.f;
    for (int i = 0; i < NN; ++i) s1 += gin[b*NN + i];
    scal[5] = s1;
  }
  __syncthreads();
  {
    const float s1 = scal[5];
    const float s0 = (float)NN - s1;
    for (int e = tid; e < NV; e += 256) {
      int i = e / NN, j = e - i*NN;
      float gi = gin[b*NN + i];
      float fi = gi / s1 - (1.f - gi) / s0;
      sw[e] = fi * vin[j];
      sp[e] = -xin[b*NV + e];
      sz[e] = 0.04f;
    }
  }
  __syncthreads();

  float t = 1.0f;
  for (int step = 0; step < NSTEPS; ++step) {
    // ---- 1) elementwise barrier gradient / Hessian diagonal ----
    for (int e = tid; e < NV; e += 256) {
      float xe = sz[e];
      float rx = 1.f / xe;
      float r1 = 1.f / (1.f - xe);
      float ge = t * (EPSQ * xe + sp[e]) - rx + r1;
      float hd = t * EPSQ + rx*rx + r1*r1;
      float de = 1.f / hd;
      sg[e]  = ge;
      sd[e]  = de;
      sdx[e] = de * ge;            // temp: d*g
    }
    __syncthreads();

    // ---- 2) structured reductions (A is row/col indicators + one dense row) ----
    {
      float pw = 0.f, pr = 0.f;
      for (int e = tid; e < NV; e += 256) {
        float de = sd[e], we = sw[e];
        pw += de * we * we;        // w' D w
        pr += sdx[e] * we;         // w' (d*g)
      }
      red[tid]       = pw;
      red[256 + tid] = pr;
    }
    if (tid < NN) {                      // row-constraint sums
      int i = tid;
      float rs = 0.f, rwl = 0.f, rr = 0.f;
      for (int j = 0; j < NN; ++j) {
        int e = i*NN + j;
        float de = sd[e];
        rs += de; rwl += de * sw[e]; rr += sdx[e];
      }
      rowsum[i] = rs; rwv[i] = rwl; rrv[i] = rr;
    } else if (tid >= 32 && tid < 32 + NN) {  // col-constraint sums
      int k = tid - 32;
      float cs = 0.f, cwl = 0.f, rc = 0.f;
      for (int i = 0; i < NN; ++i) {
        int e = i*NN + k;
        float de = sd[e];
        cs += de; cwl += de * sw[e]; rc += sdx[e];
      }
      colsum[k] = cs; cwv[k] = cwl; rcv[k] = rc;
    }
    __syncthreads();
    for (int off = 128; off > 0; off >>= 1) {
      if (tid < off) {
        red[tid]       += red[tid + off];
        red[256 + tid] += red[256 + tid + off];
      }
      __syncthreads();
    }
    if (tid == 0) {
      float sww = red[0];
      float dsum = sww;
      for (int i = 0; i < NN; ++i)     dsum += rowsum[i];
      for (int k = 0; k < NN - 1; ++k) dsum += colsum[k];
      scal[0] = 1e-7f * dsum / 50.f;   // reg = 1e-7 * mean(diag(M))
      scal[1] = sww;
      scal[2] = red[256];
    }
    __syncthreads();

    // ---- 3) build padded 64x64 normal matrix M = A D A^T + reg*I ----
    {
      float* mflat = &sM[0][0];
      for (int idx = tid; idx < MP*MST; idx += 256) mflat[idx] = 0.f;
    }
    __syncthreads();
    {
      const float reg = scal[0];
      if (tid < NN) {
        sM[tid][tid] = rowsum[tid] + reg;
        sM[tid][49]  = rwv[tid];
        sM[49][tid]  = rwv[tid];
        snu[tid]     = -rrv[tid];
      } else if (tid >= 32 && tid < 32 + NN - 1) {
        int k = tid - 32;
        sM[NN + k][NN + k] = colsum[k] + reg;
        sM[NN + k][49]     = cwv[k];
        sM[49][NN + k]     = cwv[k];
        snu[NN + k]        = -rcv[k];
      } else if (tid == 60) {
        sM[49][49] = scal[1] + reg;
        snu[49]    = -scal[2];
      } else if (tid >= 64 && tid < 64 + (MP - 50)) {
        int q = 50 + (tid - 64);       // identity padding rows, zero rhs
        sM[q][q] = 1.f;
        snu[q]   = 0.f;
      }
      for (int e = tid; e < NN*(NN-1); e += 256) {  // cross block = d[i,k]
        int i = e / (NN-1), k = e - i*(NN-1);
        float v = sd[i*NN + k];
        sM[i][NN + k] = v;
        sM[NN + k][i] = v;
      }
    }
    __syncthreads();

    // ---- 4) blocked LU (SPD + reg, no pivoting) with WMMA Schur updates ----
    for (int kb = 0; kb < 4; ++kb) {
      const int b0 = kb * 16;
      // panel factor: cols b0..b0+15, rows b0..63
      for (int k = b0; k < b0 + 16; ++k) {
        __syncthreads();
        float inv = 1.f / sM[k][k];
        if (tid < 63 - k) {
          int i = k + 1 + tid;
          float l = sM[i][k] * inv;
          sM[i][k] = l;
          for (int j = k + 1; j < b0 + 16; ++j) sM[i][j] -= l * sM[k][j];
        }
      }
      if (kb < 3) {
        const int c0  = b0 + 16;
        const int ncj = MP - c0;
        // trsm: U12 = L11^-1 A12
        for (int k = b0; k < b0 + 16; ++k) {
          __syncthreads();
          int rows = b0 + 15 - k;
          for (int u = tid; u < rows * ncj; u += 256) {
            int i = k + 1 + u / ncj;
            int j = c0 + (u - (u / ncj) * ncj);
            sM[i][j] -= sM[i][k] * sM[k][j];
          }
        }
        __syncthreads();
        // Schur: A22 -= L21 * U12  via V_WMMA_F32_16X16X4_F32, 1 tile per wave
        const int nt = ncj >> 4;
        for (int tId = wave; tId < nt * nt; tId += 8) {
          const int trow = c0 + (tId / nt) * 16;
          const int tcol = c0 + (tId % nt) * 16;
          v8f acc = {};
#pragma unroll
          for (int kk = 0; kk < 16; kk += 4) {
            int kbase = b0 + kk + ((lane >> 4) << 1);
            v2f a, bb;
            int am = trow + (lane & 15);         // A: lane holds row m, K pair
            a.x = sM[am][kbase];
            a.y = sM[am][kbase + 1];
            int bn = tcol + (lane & 15);         // B: lane holds col n, K pair
            bb.x = sM[kbase][bn];
            bb.y = sM[kbase + 1][bn];
            acc = __builtin_amdgcn_wmma_f32_16x16x4_f32(false, a, false, bb,
                                                        (short)0, acc, false, false);
          }
          int cm = trow + ((lane >> 4) << 3);    // D: vgpr v -> row v (+8 hi half)
          int cn = tcol + (lane & 15);
#pragma unroll
          for (int v = 0; v < 8; ++v) sM[cm + v][cn] -= acc[v];
        }
        __syncthreads();
      }
    }
    __syncthreads();

    // ---- 5) forward/back substitution on the real 50x50 part ----
    for (int k = 0; k < 49; ++k) {
      __syncthreads();
      if (tid < 49 - k) {
        int i = k + 1 + tid;
        snu[i] -= sM[i][k] * snu[k];
      }
    }
    __syncthreads();
    for (int k = 49; k >= 0; --k) {
      float val = snu[k] / sM[k][k];
      __syncthreads();
      if (tid == 0) snu[k] = val;
      if (tid < k) snu[tid] -= sM[tid][k] * val;
      __syncthreads();
    }

    // ---- 6) Newton direction, fraction-to-boundary step, update ----
    {
      float alo = 1e30f, ahi = 1e30f;
      float nuw = snu[49];
      for (int e = tid; e < NV; e += 256) {
        int i = e / NN, j = e - i*NN;
        float atn = snu[i] + nuw * sw[e];         // A^T nu (structured)
        if (j < NN - 1) atn += snu[NN + j];
        float dxe = -sd[e] * (sg[e] + atn);
        sdx[e] = dxe;
        float xe = sz[e];
        if (dxe < 0.f) alo = fminf(alo, -xe / dxe);
        if (dxe > 0.f) ahi = fminf(ahi, (1.f - xe) / dxe);
      }
      red[tid]       = alo;
      red[256 + tid] = ahi;
    }
    __syncthreads();
    for (int off = 128; off > 0; off >>= 1) {
      if (tid < off) {
        red[tid]       = fminf(red[tid],       red[tid + off]);
        red[256 + tid] = fminf(red[256 + tid], red[256 + tid + off]);
      }
      __syncthreads();
    }
    if (tid == 0) scal[3] = fminf(1.f, 0.99f * fminf(red[0], red[256]));
    __syncthreads();
    {
      const float alpha = scal[3];
      for (int e = tid; e < NV; e += 256) sz[e] += alpha * sdx[e];
    }
    if (step % 6 == 5) t *= 10.f;   // ts = repeat(MU^arange(8), 6)
    __syncthreads();
  }

  for (int e = tid; e < NV; e += 256) out[b*NV + e] = sz[e];
}

extern "C" void kernel_launch(void* const* d_in, const int* in_sizes, int n_in,
                              void* d_out, int out_size, void* d_ws, size_t ws_size,
                              hipStream_t stream) {
  const float* x = (const float*)d_in[0];   // [B, 625] scores
  const float* g = (const float*)d_in[1];   // [B, 25] group ids
  const float* v = (const float*)d_in[2];   // [25] exposure
  float* o = (float*)d_out;                 // [B, 25, 25]
  const int B = in_sizes[0] / NV;
  policy_lp_kernel<<<B, 256, 0, stream>>>(x, g, v, o);
}